// Layer_VanillaTransformer_16535624089993
// MI455X (gfx1250) — compile-verified
//
#include <hip/hip_runtime.h>
#include <hip/hip_bf16.h>

typedef float v2f __attribute__((ext_vector_type(2)));
typedef float v8f __attribute__((ext_vector_type(8)));

#define N_NODES   4096
#define IN_DIM    512
#define REPR_DIM  512
#define HID_DIM   2048
#define LN_EPS    1e-5f

// ---------------------------------------------------------------------------
// K1: wv[h,m] = sum_k Wvs[h,k,m]   (8 x 512 x 64 -> 512 values)
// ---------------------------------------------------------------------------
__global__ void prep_wv_kernel(const float* __restrict__ Wvs, float* __restrict__ wv) {
    const int p = threadIdx.x;            // 512 threads, p = h*64 + m
    const int head = p >> 6;
    const int m = p & 63;
    const float* src = Wvs + (size_t)head * (512 * 64) + m;
    float s = 0.0f;
#pragma unroll 8
    for (int k = 0; k < 512; ++k) s += src[k * 64];
    wv[p] = s;
}

// ---------------------------------------------------------------------------
// K2: partial sums of node_matrix (2,097,152 floats) -> part[512]
// 512 blocks x 256 threads, float4, deterministic tree reduction
// ---------------------------------------------------------------------------
__global__ void nodesum_kernel(const float4* __restrict__ in4, float* __restrict__ part) {
    __shared__ float red[256];
    const int tid = threadIdx.x;
    const int idx = blockIdx.x * 256 + tid;       // 131072 threads total
    float s = 0.0f;
#pragma unroll
    for (int r = 0; r < 4; ++r) {
        float4 v = in4[idx + r * 131072];         // 524288 float4 total
        s += (v.x + v.y) + (v.z + v.w);
    }
    red[tid] = s;
    __syncthreads();
    for (int st = 128; st > 0; st >>= 1) {
        if (tid < st) red[tid] += red[tid + st];
        __syncthreads();
    }
    if (tid == 0) part[blockIdx.x] = red[0];
}

// ---------------------------------------------------------------------------
// K3/K5/K6: GEMV via V_WMMA_F32_16X16X4_F32 with broadcast-A rows.
// out[j] = (relu?)( sum_k x[k]*W[k,j] + bias[j] )
// Each wave owns 16 consecutive output columns; K accumulated in steps of 4.
// A 16x4 f32 layout (ISA 7.12.2): lane<16 -> K=k,k+1 ; lane>=16 -> K=k+2,k+3.
// B 4x16 mirrors it: reg v, lane l -> (K = k + v + 2*(l>>4), N = j0 + (l&15)).
// D 16x16: all rows equal; lane<16 reg0 = row M=0 -> write result.
// ---------------------------------------------------------------------------
__global__ void gemv_wmma_kernel(const float* __restrict__ x,
                                 const float* __restrict__ W,
                                 const float* __restrict__ bias,
                                 float* __restrict__ out,
                                 int K, int N, int do_relu)
{
    __shared__ float xs[2048];
    const int tid = threadIdx.x;
    for (int i = tid; i < K; i += blockDim.x) xs[i] = x[i];
    __syncthreads();

    const int wave = (blockIdx.x * blockDim.x + tid) >> 5;  // global wave id
    const int lane = tid & 31;
    const int half = lane >> 4;          // 0 | 1
    const int nidx = lane & 15;
    const int j0   = wave * 16;          // launch sized so j0 < N always

    const float* Wcol = W + (size_t)j0 + nidx;

    v8f acc = {0.f, 0.f, 0.f, 0.f, 0.f, 0.f, 0.f, 0.f};
#pragma unroll 2
    for (int k = 0; k < K; k += 4) {
        const int ka = k + 2 * half;
        v2f a, b;
        a.x = xs[ka];
        a.y = xs[ka + 1];
        b.x = Wcol[(size_t)ka * N];
        b.y = Wcol[(size_t)(ka + 1) * N];
        acc = __builtin_amdgcn_wmma_f32_16x16x4_f32(
                  /*neg_a=*/false, a, /*neg_b=*/false, b,
                  /*c_mod=*/(short)0, acc, /*reuse_a=*/false, /*reuse_b=*/false);
    }

    if (half == 0) {                     // lanes 0..15 hold row M=0 in acc[0]
        float v = acc[0];
        if (bias) v += bias[j0 + nidx];
        if (do_relu) v = fmaxf(v, 0.0f);
        out[j0 + nidx] = v;
    }
}

// ---------------------------------------------------------------------------
// K4: S = sum(part[512]); x = S*t; LayerNorm(g1,be1) -> out  (1 block, 512 thr)
// ---------------------------------------------------------------------------
__global__ void ln1_kernel(const float* __restrict__ part, const float* __restrict__ t,
                           const float* __restrict__ gamma, const float* __restrict__ beta,
                           float* __restrict__ out)
{
    __shared__ float red[512];
    __shared__ float sS, smu, srstd;
    const int tid = threadIdx.x;

    red[tid] = part[tid];
    __syncthreads();
    for (int s = 256; s > 0; s >>= 1) { if (tid < s) red[tid] += red[tid + s]; __syncthreads(); }
    if (tid == 0) sS = red[0];
    __syncthreads();

    const float x = sS * t[tid];
    red[tid] = x;
    __syncthreads();
    for (int s = 256; s > 0; s >>= 1) { if (tid < s) red[tid] += red[tid + s]; __syncthreads(); }
    if (tid == 0) smu = red[0] * (1.0f / 512.0f);
    __syncthreads();

    const float d = x - smu;
    red[tid] = d * d;
    __syncthreads();
    for (int s = 256; s > 0; s >>= 1) { if (tid < s) red[tid] += red[tid + s]; __syncthreads(); }
    if (tid == 0) srstd = rsqrtf(red[0] * (1.0f / 512.0f) + LN_EPS);
    __syncthreads();

    out[tid] = d * srstd * gamma[tid] + beta[tid];
}

// ---------------------------------------------------------------------------
// K7: plain LayerNorm(z; g2,be2) -> r   (1 block, 512 thr)
// ---------------------------------------------------------------------------
__global__ void ln2_kernel(const float* __restrict__ z,
                           const float* __restrict__ gamma, const float* __restrict__ beta,
                           float* __restrict__ out)
{
    __shared__ float red[512];
    __shared__ float smu, srstd;
    const int tid = threadIdx.x;

    const float x = z[tid];
    red[tid] = x;
    __syncthreads();
    for (int s = 256; s > 0; s >>= 1) { if (tid < s) red[tid] += red[tid + s]; __syncthreads(); }
    if (tid == 0) smu = red[0] * (1.0f / 512.0f);
    __syncthreads();

    const float d = x - smu;
    red[tid] = d * d;
    __syncthreads();
    for (int s = 256; s > 0; s >>= 1) { if (tid < s) red[tid] += red[tid + s]; __syncthreads(); }
    if (tid == 0) srstd = rsqrtf(red[0] * (1.0f / 512.0f) + LN_EPS);
    __syncthreads();

    out[tid] = d * srstd * gamma[tid] + beta[tid];
}

// ---------------------------------------------------------------------------
// K8: broadcast row r[512] to all 4096 output rows (8 MB store, float4)
// ---------------------------------------------------------------------------
__global__ void bcast_kernel(const float4* __restrict__ r4, float4* __restrict__ out4) {
    const int row = blockIdx.x;          // 4096
    const int c = threadIdx.x;           // 128 (512 floats / 4)
    out4[(size_t)row * 128 + c] = r4[c];
}

// ---------------------------------------------------------------------------
extern "C" void kernel_launch(void* const* d_in, const int* in_sizes, int n_in,
                              void* d_out, int out_size, void* d_ws, size_t ws_size,
                              hipStream_t stream) {
    const float* node = (const float*)d_in[0];
    // d_in[1] adjacency, d_in[2] Wks, d_in[3] Wqs: provably unused by the math.
    const float* Wvs = (const float*)d_in[4];
    const float* Wo  = (const float*)d_in[5];
    const float* W1  = (const float*)d_in[6];
    const float* b1  = (const float*)d_in[7];
    const float* W2  = (const float*)d_in[8];
    const float* b2  = (const float*)d_in[9];
    const float* g1  = (const float*)d_in[10];
    const float* be1 = (const float*)d_in[11];
    const float* g2  = (const float*)d_in[12];
    const float* be2 = (const float*)d_in[13];
    float* out = (float*)d_out;

    float* wsf  = (float*)d_ws;
    float* part = wsf;            // [512]  per-block partial node sums
    float* wv   = wsf + 512;      // [512]  colsum(Wvs) flattened (h,m)
    float* t    = wsf + 1024;     // [512]  wv . Wo
    float* xn   = wsf + 1536;     // [512]  after LN1
    float* h    = wsf + 2048;     // [2048] hidden (post-ReLU)
    float* z    = wsf + 4096;     // [512]  FFN output
    float* r    = wsf + 4608;     // [512]  final row after LN2

    prep_wv_kernel  <<<1,    512, 0, stream>>>(Wvs, wv);
    nodesum_kernel  <<<512,  256, 0, stream>>>((const float4*)node, part);
    // t[n] = sum_p wv[p] * Wo_flat[p,n]   (Wo is (8,64,512) contiguous = 512x512)
    gemv_wmma_kernel<<<4,    256, 0, stream>>>(wv, Wo, nullptr, t, 512, 512, 0);
    ln1_kernel      <<<1,    512, 0, stream>>>(part, t, g1, be1, xn);
    gemv_wmma_kernel<<<16,   256, 0, stream>>>(xn, W1, b1, h, 512, 2048, 1);
    gemv_wmma_kernel<<<4,    256, 0, stream>>>(h,  W2, b2, z, 2048, 512, 0);
    ln2_kernel      <<<1,    512, 0, stream>>>(z, g2, be2, r);
    bcast_kernel    <<<4096, 128, 0, stream>>>((const float4*)r, (float4*)out);
}